// RD_loss5_90331752169941
// MI455X (gfx1250) — compile-verified
//
#include <hip/hip_runtime.h>

typedef float v2f __attribute__((ext_vector_type(2)));
typedef float v8f __attribute__((ext_vector_type(8)));

#define HH 384
#define WW 384
#define PN 1000
#define BN 64

// Exact 32-lane all-reduce (sum) using V_WMMA_F32_16X16X4_F32.
// Step 1: D1 = A(v in VGPR0, 0 in VGPR1) x B(all ones):
//   A layout (16x4 f32): lane m -> A[m][0], lane m+16 -> A[m][2]
//   => D1[m][n] = v_m + v_{m+16}  (rowsum_m, constant over n)
//   D layout: c1[r] = rowsum_r (lanes 0-15) / rowsum_{r+8} (lanes 16-31),
//   which is exactly the 4x16 f32 B-matrix layout.
// Steps 2-5: acc += A(ones) x B(c1[2j], c1[2j+1]) adds
//   rowsum_{2j}+rowsum_{2j+1}+rowsum_{2j+8}+rowsum_{2j+9}; j=0..3 covers all 16.
// Every element of acc ends up = sum over all 32 lanes of v.
__device__ __forceinline__ float wave32_sum_wmma(float v) {
    v2f a;    a[0] = v;    a[1] = 0.0f;
    v2f ones; ones[0] = 1.0f; ones[1] = 1.0f;

    v8f c1 = {};
    c1 = __builtin_amdgcn_wmma_f32_16x16x4_f32(false, a, false, ones,
                                               (short)0, c1, false, false);
    v8f acc = {};
    v2f b;
    b[0] = c1[0]; b[1] = c1[1];
    acc = __builtin_amdgcn_wmma_f32_16x16x4_f32(false, ones, false, b,
                                                (short)0, acc, false, false);
    b[0] = c1[2]; b[1] = c1[3];
    acc = __builtin_amdgcn_wmma_f32_16x16x4_f32(false, ones, false, b,
                                                (short)0, acc, false, false);
    b[0] = c1[4]; b[1] = c1[5];
    acc = __builtin_amdgcn_wmma_f32_16x16x4_f32(false, ones, false, b,
                                                (short)0, acc, false, false);
    b[0] = c1[6]; b[1] = c1[7];
    acc = __builtin_amdgcn_wmma_f32_16x16x4_f32(false, ones, false, b,
                                                (short)0, acc, false, false);
    return acc[0];
}

__global__ void rd_loss_partial(const float* __restrict__ pred,
                                const float* __restrict__ gt,
                                const int*   __restrict__ px,
                                const int*   __restrict__ py,
                                float*       __restrict__ partial) {
    __shared__ float wave_sums[8];

    const int p = blockIdx.x * blockDim.x + threadIdx.x;   // sample point
    const int b = blockIdx.y;                              // batch image

    float t = 0.0f;
    if (p < PN) {
        const int x = px[p];
        const int y = py[p];
        const float* __restrict__ dp = pred + (size_t)b * (HH * WW);
        const float* __restrict__ dg = gt   + (size_t)b * (HH * WW);
        const int c = y * WW + x;
        // Neighbor offsets matching _DY/_DX order:
        // dy = {-1,1,0,0,-1,-1,1,1}, dx = {0,0,-1,1,-1,1,-1,1}
        const int off[8] = { -WW, WW, -1, 1, -WW - 1, -WW + 1, WW - 1, WW + 1 };

        const float cp = dp[c];
        const float cg = dg[c];
        float vp[8], vg[8];
        float np = 0.0f, ng = 0.0f;
#pragma unroll
        for (int i = 0; i < 8; ++i) {
            vp[i] = cp - dp[c + off[i]];
            vg[i] = cg - dg[c + off[i]];
            np += vp[i] * vp[i];
            ng += vg[i] * vg[i];
        }
        np = sqrtf(np);
        ng = sqrtf(ng);
        const float inp = (np == 0.0f) ? 1.0f : 1.0f / np;
        const float ing = (ng == 0.0f) ? 1.0f : 1.0f / ng;
#pragma unroll
        for (int i = 0; i < 8; ++i)
            t += fabsf(vp[i] * inp - vg[i] * ing);
    }

    // All 32 lanes of every wave reach here (EXEC all-1s) -> WMMA legal.
    const float wsum = wave32_sum_wmma(t);

    const int lane = threadIdx.x & 31;
    const int wid  = threadIdx.x >> 5;
    if (lane == 0) wave_sums[wid] = wsum;
    __syncthreads();
    if (threadIdx.x == 0) {
        float s = 0.0f;
#pragma unroll
        for (int i = 0; i < 8; ++i) s += wave_sums[i];
        partial[blockIdx.y * gridDim.x + blockIdx.x] = s;
    }
}

__global__ void rd_loss_final(const float* __restrict__ partial,
                              float* __restrict__ out,
                              int n, float scale) {
    __shared__ float s[256];
    s[threadIdx.x] = (threadIdx.x < n) ? partial[threadIdx.x] : 0.0f;
    __syncthreads();
#pragma unroll
    for (int stride = 128; stride > 0; stride >>= 1) {
        if ((int)threadIdx.x < stride) s[threadIdx.x] += s[threadIdx.x + stride];
        __syncthreads();
    }
    if (threadIdx.x == 0) out[0] = s[0] * scale;
}

extern "C" void kernel_launch(void* const* d_in, const int* in_sizes, int n_in,
                              void* d_out, int out_size, void* d_ws, size_t ws_size,
                              hipStream_t stream) {
    (void)in_sizes; (void)n_in; (void)out_size; (void)ws_size;
    const float* pred = (const float*)d_in[0];   // (64,1,384,384) f32
    const float* gt   = (const float*)d_in[1];   // (64,1,384,384) f32
    const int*   px   = (const int*)d_in[2];     // (1000,) i32
    const int*   py   = (const int*)d_in[3];     // (1000,) i32
    float* partial = (float*)d_ws;               // 256 floats of scratch
    float* out     = (float*)d_out;              // scalar f32

    dim3 grid((PN + 255) / 256, BN);             // (4, 64) blocks of 256
    rd_loss_partial<<<grid, 256, 0, stream>>>(pred, gt, px, py, partial);

    const int nPartial = grid.x * BN;            // 256
    const float scale = 1.0f / (float)(BN * PN * 8);
    rd_loss_final<<<1, 256, 0, stream>>>(partial, out, nPartial, scale);
}